// PillarScatter_56650618634919
// MI455X (gfx1250) — compile-verified
//
#include <hip/hip_runtime.h>
#include <cstdint>
#include <cstddef>

typedef float vfloat4 __attribute__((ext_vector_type(4)));
typedef int   vint4   __attribute__((ext_vector_type(4)));

#define NX_G   512
#define NY_G   512
#define NXY_G  (NX_G * NY_G)   // 262144 cells per batch
#define B_G    4
#define P_G    12000
#define C_G    64
#define TILE_G 1024            // cells per workgroup in the gather kernel
#define TPB_G  256             // 8 waves (wave32)

// ---------------- Kernel: gather + streaming NT write ----------------------
// One workgroup = one 1024-cell tile of one batch. The 4 KB map tile is
// staged into LDS with the CDNA5 async copy path (ASYNCcnt), read back with
// ds_load_b128, then 64 channel planes are written as coalesced non-temporal
// b128 stores.
__global__ __launch_bounds__(TPB_G)
void pscat_gather(const float* __restrict__ emb,
                  const int* __restrict__ map,
                  float* __restrict__ out) {
  __shared__ int smap[TILE_G];

  const int b         = blockIdx.y;
  const int cell_base = blockIdx.x * TILE_G;
  const int tid       = threadIdx.x;

  // LDS byte offset of this lane's 16B slot. On AMDGCN a generic pointer to
  // LDS carries the LDS byte address in its low 32 bits;
  // this also keeps the LDS allocation alive.
  const unsigned lds_w = (unsigned)(size_t)(&smap[tid * 4]);

  // Async-stage the 4 KB map tile into LDS: 256 lanes x 16 B.
  {
    const int* gsrc = map + (size_t)b * NXY_G + cell_base + tid * 4;
    asm volatile("global_load_async_to_lds_b128 %0, %1, off"
                 :: "v"(lds_w), "v"(gsrc)
                 : "memory");
  }
  asm volatile("s_wait_asynccnt 0" ::: "memory");
  __syncthreads();

  // Read back this lane's 4 cell indices from LDS. The s_wait_dscnt is fused
  // into the same asm statement so the output registers cannot be consumed
  // by scheduled code before the DS return lands (DScnt==0).
  vint4 p4;
  asm volatile("ds_load_b128 %0, %1\n\t"
               "s_wait_dscnt 0"
               : "=v"(p4)
               : "v"(lds_w)
               : "memory");

  const int ci = tid * 4;  // this thread's 4 consecutive cells
  float* obase = out + ((size_t)b * C_G) * NXY_G + (size_t)(cell_base + ci);

  // Fast path: all 4 cells empty (~91% of groups) -> pure zero streaming.
  if ((p4.x & p4.y & p4.z & p4.w) < 0) {
    vfloat4 z = { 0.f, 0.f, 0.f, 0.f };
#pragma unroll 8
    for (int c = 0; c < C_G; ++c)
      __builtin_nontemporal_store(
          z, reinterpret_cast<vfloat4*>(obase + (size_t)c * NXY_G));
    return;
  }

  // General path: per-lane sparse gather out of L2-resident embeddings.
  const float* eb = emb + (size_t)b * P_G * C_G;
  const int o0 = p4.x * C_G;
  const int o1 = p4.y * C_G;
  const int o2 = p4.z * C_G;
  const int o3 = p4.w * C_G;
#pragma unroll 4
  for (int c = 0; c < C_G; ++c) {
    vfloat4 v;
    v.x = (p4.x >= 0) ? eb[o0 + c] : 0.f;
    v.y = (p4.y >= 0) ? eb[o1 + c] : 0.f;
    v.z = (p4.z >= 0) ? eb[o2 + c] : 0.f;
    v.w = (p4.w >= 0) ? eb[o3 + c] : 0.f;
    __builtin_nontemporal_store(
        v, reinterpret_cast<vfloat4*>(obase + (size_t)c * NXY_G));
  }
}

// ---------------- Kernel: init cell->pillar map to -1 ----------------------
// 4 MB map lives in d_ws; regular (RT) stores so it stays resident in the
// 192 MB L2 for the following two kernels.
__global__ __launch_bounds__(TPB_G)
void pscat_init_map(int* __restrict__ map, int n4) {
  int i = blockIdx.x * blockDim.x + threadIdx.x;
  if (i < n4) {
    vint4 m = { -1, -1, -1, -1 };
    reinterpret_cast<vint4*>(map)[i] = m;
  }
}

// ---------------- Kernel: scatter pillar index into the map ----------------
// Coordinates are unique per batch (voxelization guarantee) -> race-free.
__global__ __launch_bounds__(TPB_G)
void pscat_scatter_idx(const int* __restrict__ coords,
                       const int* __restrict__ mask,
                       int* __restrict__ map) {
  int t = blockIdx.x * blockDim.x + threadIdx.x;
  if (t >= B_G * P_G) return;
  if (mask[t] > 0) {
    int b = t / P_G;
    int p = t - b * P_G;
    int2 c2 = reinterpret_cast<const int2*>(coords)[t];  // (ix, iy)
    if ((unsigned)c2.x < (unsigned)NX_G && (unsigned)c2.y < (unsigned)NY_G)
      map[b * NXY_G + c2.y * NX_G + c2.x] = p;
  }
}

extern "C" void kernel_launch(void* const* d_in, const int* in_sizes, int n_in,
                              void* d_out, int out_size, void* d_ws, size_t ws_size,
                              hipStream_t stream) {
  (void)in_sizes; (void)n_in; (void)out_size; (void)ws_size;

  const float* emb    = (const float*)d_in[0];  // [B, P, C] fp32
  const int*   coords = (const int*)d_in[1];    // [B, P, 2] int32 (ix, iy)
  const int*   mask   = (const int*)d_in[2];    // [B, P]    int32
  float*       out    = (float*)d_out;          // [B, C, NY, NX] fp32
  int*         map    = (int*)d_ws;             // 4 * 262144 * 4 B = 4 MB

  const int n4 = B_G * NXY_G / 4;               // int4 fill count
  pscat_init_map<<<dim3((n4 + TPB_G - 1) / TPB_G), dim3(TPB_G), 0, stream>>>(map, n4);

  const int nt = B_G * P_G;
  pscat_scatter_idx<<<dim3((nt + TPB_G - 1) / TPB_G), dim3(TPB_G), 0, stream>>>(
      coords, mask, map);

  pscat_gather<<<dim3(NXY_G / TILE_G, B_G), dim3(TPB_G), 0, stream>>>(emb, map, out);
}